// RayCastLayerV2_5463198400793
// MI455X (gfx1250) — compile-verified
//
#include <hip/hip_runtime.h>

typedef __attribute__((ext_vector_type(2))) float v2f;
typedef __attribute__((ext_vector_type(8))) float v8f;

#define BOARD 19
#define HW    361
#define CIN   32
#define COUT  64
#define MD    18
#define K2    64      // 2*CIN : [SL channels ; SD channels]
#define NPAD  368     // 361 padded up to multiple of 16
#define NDIAG 37

// LDS partition (offsets in floats)
#define OFF_X    0                        // x[b]      : 32*361 = 11552
#define OFF_RS   (OFF_X  + CIN*HW)        // rowsum    : 32*19
#define OFF_CS   (OFF_RS + CIN*BOARD)     // colsum    : 32*19
#define OFF_DG   (OFF_CS + CIN*BOARD)     // diagsum   : 32*37
#define OFF_AD   (OFF_DG + CIN*NDIAG)     // antisum   : 32*37
#define OFF_S2   (OFF_AD + CIN*NDIAG)     // S2[k][n]  : 64*368 = 23552
#define OFF_W2   (OFF_S2 + K2*NPAD)       // W2[o][k]  : 64*64  = 4096
#define OFF_BIAS (OFF_W2 + COUT*K2)       // bias      : 64
#define LDS_FLOATS (OFF_BIAS + COUT)      // = 42848 floats = 171392 B (< 320 KB/WGP)

// ---- prep: W2[o][k] = sum_R weight[o, c, g, R]   (k<32: g=0,c=k ; k>=32: g=1,c=k-32)
__global__ void raycast_prep_w2(const float* __restrict__ w, float* __restrict__ w2) {
  int i = blockIdx.x * blockDim.x + threadIdx.x;
  if (i >= COUT * K2) return;
  int o = i >> 6;
  int k = i & 63;
  int g = (k >= CIN) ? 1 : 0;
  int c = k - g * CIN;
  const float* p = w + (((o * CIN + c) * 2 + g) * MD);
  float s = 0.f;
#pragma unroll
  for (int r = 0; r < MD; ++r) s += p[r];
  w2[o * K2 + k] = s;
}

// ---- fused main: ray sums (analytic) + 64x64xN GEMM via V_WMMA_F32_16X16X4_F32
__global__ void __launch_bounds__(256, 1)
raycast_main(const float* __restrict__ x, const float* __restrict__ w2g,
             const float* __restrict__ bias, float* __restrict__ out) {
  extern __shared__ float smem[];
  const int tid = threadIdx.x;
  const int b   = blockIdx.x;
  const float* xb = x + (size_t)b * CIN * HW;

  // Stage 0: stage x, W2, bias into LDS (coalesced)
  for (int i = tid; i < CIN * HW;  i += 256) smem[OFF_X  + i] = xb[i];
  for (int i = tid; i < COUT * K2; i += 256) smem[OFF_W2 + i] = w2g[i];
  if (tid < COUT) smem[OFF_BIAS + tid] = bias[tid];
  __syncthreads();

  // Stage 1: per-channel directional totals (no atomics; 3584 independent tasks)
  for (int t = tid; t < 3584; t += 256) {
    if (t < 608) {                                  // rowsum[c][h]
      int c = t / BOARD, h = t % BOARD;
      const float* row = &smem[OFF_X + c * HW + h * BOARD];
      float s = 0.f;
#pragma unroll
      for (int w = 0; w < BOARD; ++w) s += row[w];
      smem[OFF_RS + c * BOARD + h] = s;
    } else if (t < 1216) {                          // colsum[c][w]
      int u = t - 608; int c = u / BOARD, w = u % BOARD;
      const float* base = &smem[OFF_X + c * HW + w];
      float s = 0.f;
#pragma unroll
      for (int h = 0; h < BOARD; ++h) s += base[h * BOARD];
      smem[OFF_CS + c * BOARD + w] = s;
    } else if (t < 2400) {                          // diagsum[c][h-w+18]
      int u = t - 1216; int c = u / NDIAG, d = u % NDIAG;
      float s = 0.f;
      for (int h = 0; h < BOARD; ++h) {
        int w = h - (d - 18);
        if ((unsigned)w < BOARD) s += smem[OFF_X + c * HW + h * BOARD + w];
      }
      smem[OFF_DG + c * NDIAG + d] = s;
    } else {                                        // antisum[c][h+w]
      int u = t - 2400; int c = u / NDIAG, a = u % NDIAG;
      float s = 0.f;
      for (int h = 0; h < BOARD; ++h) {
        int w = a - h;
        if ((unsigned)w < BOARD) s += smem[OFF_X + c * HW + h * BOARD + w];
      }
      smem[OFF_AD + c * NDIAG + a] = s;
    }
  }
  __syncthreads();

  // Stage 2: S2[k][n]  (SL = rowsum+colsum-2x ; SD = diag+anti-2x)
  for (int i = tid; i < K2 * HW; i += 256) {
    int k = i / HW, n = i - k * HW;
    int h = n / BOARD, w = n - h * BOARD;
    float v;
    if (k < CIN) {
      float xv = smem[OFF_X + k * HW + n];
      v = smem[OFF_RS + k * BOARD + h] + smem[OFF_CS + k * BOARD + w] - 2.f * xv;
    } else {
      int c = k - CIN;
      float xv = smem[OFF_X + c * HW + n];
      v = smem[OFF_DG + c * NDIAG + (h - w + 18)] +
          smem[OFF_AD + c * NDIAG + (h + w)] - 2.f * xv;
    }
    smem[OFF_S2 + k * NPAD + n] = v;
  }
  for (int i = tid; i < K2 * (NPAD - HW); i += 256) {   // zero padded N columns
    int k = i / (NPAD - HW), n = HW + i % (NPAD - HW);
    smem[OFF_S2 + k * NPAD + n] = 0.f;
  }
  __syncthreads();

  // Stage 3: out[b, 0:64, n] = W2[64x64] * S2[64 x NPAD]  (f32 WMMA, K=64 = 16 steps of 4)
  const int lane = tid & 31;
  const int wave = tid >> 5;
  const int lr   = lane & 15;
  const int koff = (lane >> 4) << 1;          // lanes 16-31 carry K+2, K+3
  const int NT = NPAD / 16;                   // 23 N-tiles
  for (int tile = wave; tile < (COUT / 16) * NT; tile += 8) {
    const int mt = tile / NT;
    const int nt = tile - mt * NT;
    const int m0 = mt * 16, n0 = nt * 16;
    v8f acc = {};
#pragma unroll
    for (int kk = 0; kk < 16; ++kk) {
      const int k0 = kk * 4 + koff;
      v2f a, bb;
      a.x  = smem[OFF_W2 + (m0 + lr) * K2 + k0];
      a.y  = smem[OFF_W2 + (m0 + lr) * K2 + k0 + 1];
      bb.x = smem[OFF_S2 + (k0    ) * NPAD + n0 + lr];
      bb.y = smem[OFF_S2 + (k0 + 1) * NPAD + n0 + lr];
      acc = __builtin_amdgcn_wmma_f32_16x16x4_f32(
          /*neg_a=*/false, a, /*neg_b=*/false, bb,
          /*c_mod=*/(short)0, acc, /*reuse_a=*/false, /*reuse_b=*/false);
    }
    const int n = n0 + lr;
    if (n < HW) {
      const int obase = m0 + ((lane >> 4) << 3);   // lanes 16-31 hold rows M+8..M+15
      float* op = out + (size_t)b * COUT * HW + n;
#pragma unroll
      for (int j = 0; j < 8; ++j) {
        int o = obase + j;
        op[o * HW] = acc[j] + smem[OFF_BIAS + o];
      }
    }
  }
}

extern "C" void kernel_launch(void* const* d_in, const int* in_sizes, int n_in,
                              void* d_out, int out_size, void* d_ws, size_t ws_size,
                              hipStream_t stream) {
  const float* x    = (const float*)d_in[0];   // [128,32,19,19]
  const float* w    = (const float*)d_in[1];   // [64,32,2,18]
  const float* bias = (const float*)d_in[2];   // [64]
  // d_in[3] flat_index / d_in[4] valid_mask are provably redundant (masks == board bounds)
  float* out = (float*)d_out;                  // [128,64,19,19]
  float* w2  = (float*)d_ws;                   // 64*64 floats = 16 KB scratch

  raycast_prep_w2<<<16, 256, 0, stream>>>(w, w2);
  raycast_main<<<128, 256, LDS_FLOATS * sizeof(float), stream>>>(x, w2, bias, out);
}